// DynamicsEquation_31705448579547
// MI455X (gfx1250) — compile-verified
//
#include <hip/hip_runtime.h>
#include <hip/hip_bf16.h>

#define DIM 8

// Kernel: one lane per edge (grid-stride).
//   m = w_e * (x[row_e] - x[col_e]);  out[col_e] += m  (8x fp32 atomics into L2)
// Edge stream (row/col/w) is read once -> non-temporal loads + prefetch ahead.
// Node table (3.2 MB) stays L2-resident for the random gathers.
// (Placed first in the file so the disasm snippet shows the atomic lowering.)
__global__ __launch_bounds__(256) void edge_scatter_kernel(const float* __restrict__ x,
                                                           const int* __restrict__ row,
                                                           const int* __restrict__ col,
                                                           const float* __restrict__ w,
                                                           float* __restrict__ out,
                                                           int E) {
    const int stride = gridDim.x * blockDim.x;
    for (int e = blockIdx.x * blockDim.x + threadIdx.x; e < E; e += stride) {
        // Prefetch next grid-stride iteration of the edge stream
        // (speculative global_prefetch_b8: dropped if OOB, no counter cost).
        __builtin_prefetch(row + e + stride, 0, 1);
        __builtin_prefetch(col + e + stride, 0, 1);
        __builtin_prefetch(w   + e + stride, 0, 1);

        const int   r  = __builtin_nontemporal_load(row + e);
        const int   c  = __builtin_nontemporal_load(col + e);
        const float wt = __builtin_nontemporal_load(w + e);

        const float4* xr = reinterpret_cast<const float4*>(x + (size_t)r * DIM);
        const float4* xc = reinterpret_cast<const float4*>(x + (size_t)c * DIM);
        const float4 a0 = xr[0];
        const float4 a1 = xr[1];
        const float4 b0 = xc[0];
        const float4 b1 = xc[1];

        float* o = out + (size_t)c * DIM;
        atomicAdd(o + 0, (a0.x - b0.x) * wt);
        atomicAdd(o + 1, (a0.y - b0.y) * wt);
        atomicAdd(o + 2, (a0.z - b0.z) * wt);
        atomicAdd(o + 3, (a0.w - b0.w) * wt);
        atomicAdd(o + 4, (a1.x - b1.x) * wt);
        atomicAdd(o + 5, (a1.y - b1.y) * wt);
        atomicAdd(o + 6, (a1.z - b1.z) * wt);
        atomicAdd(o + 7, (a1.w - b1.w) * wt);
    }
}

// Kernel: out = -x, vectorized as float4 (global_load_b128 / global_store_b128).
// Fully rewrites d_out every call -> deterministic across graph replays.
__global__ __launch_bounds__(256) void self_term_kernel(const float* __restrict__ x,
                                                        float* __restrict__ out,
                                                        int n4) {
    int i = blockIdx.x * blockDim.x + threadIdx.x;
    if (i < n4) {
        float4 v = reinterpret_cast<const float4*>(x)[i];
        float4 o;
        o.x = -v.x; o.y = -v.y; o.z = -v.z; o.w = -v.w;
        reinterpret_cast<float4*>(out)[i] = o;
    }
}

extern "C" void kernel_launch(void* const* d_in, const int* in_sizes, int n_in,
                              void* d_out, int out_size, void* d_ws, size_t ws_size,
                              hipStream_t stream) {
    const float* state = (const float*)d_in[0];   // [N, 8] fp32
    const int*   row   = (const int*)d_in[1];     // [E] int
    const int*   col   = (const int*)d_in[2];     // [E] int
    const float* wgt   = (const float*)d_in[3];   // [E] fp32
    float*       out   = (float*)d_out;           // [N, 8] fp32

    const int nElems = in_sizes[0];               // N * 8 = 800000
    const int E      = in_sizes[1];               // 6.4M

    // Pass 1: out = -x  (must complete before atomics; same-stream ordering).
    const int n4 = nElems / 4;                    // 200000 float4s
    self_term_kernel<<<(n4 + 255) / 256, 256, 0, stream>>>(state, out, n4);

    // Pass 2: grid-stride edge scatter. 4096 blocks x 256 threads = 1M lanes
    // -> ~6 edges/lane, plenty of waves in flight to hide L2/atomic latency.
    const int blocks = 4096;
    edge_scatter_kernel<<<blocks, 256, 0, stream>>>(state, row, col, wgt, out, E);
}